// ScorePredictor_47373489275210
// MI455X (gfx1250) — compile-verified
//
#include <hip/hip_runtime.h>
#include <hip/hip_bf16.h>

typedef __attribute__((ext_vector_type(2))) float v2f;
typedef __attribute__((ext_vector_type(8))) float v8f;

// One wave32 computes 16 edge scores via the WMMA-diagonal trick:
//   D(16x16) = U(16x64) x V^T(64x16), accumulated as 16 chained
//   V_WMMA_F32_16X16X4_F32 steps; diag(D)[i] = u_i . v_i.
// f32 A fragment layout (ISA 7.12.2): lane(0..15)=M, lanes 16..31 carry K=2,3;
// VGPR0/1 = K pair => each lane loads float2 at column 4k + 2*(lane>=16)
// of its gathered row. B is symmetric with N across lanes 0..15.
__global__ __launch_bounds__(256) void edge_dot_wmma_kernel(
    const float* __restrict__ x,          // [N_NODES, 64]
    const long long* __restrict__ src,    // [E]
    const long long* __restrict__ dst,    // [E]
    float* __restrict__ out,              // [E]
    int n_edges) {
  const int lane = threadIdx.x & 31;
  const int wave = threadIdx.x >> 5;
  const int edge_base = (blockIdx.x * 8 + wave) * 16;

  if (edge_base >= n_edges) return;

  if (edge_base + 16 <= n_edges) {
    // ---- main path: full 16-edge tile, EXEC all ones (WMMA-safe) ----
    const int e = edge_base + (lane & 15);
    const long long srow = src[e];
    const long long drow = dst[e];
    const int colk = (lane >> 4) << 1;  // 0 for lanes 0-15, 2 for lanes 16-31

    const float* __restrict__ uP = x + (size_t)srow * 64 + colk;
    const float* __restrict__ vP = x + (size_t)drow * 64 + colk;

    // Issue all 32 gathers (global_load_b64) before the WMMA chain:
    // max memory-level parallelism; this kernel is L2-gather bound.
    v2f a[16], b[16];
#pragma unroll
    for (int k = 0; k < 16; ++k) {
      a[k] = *(const v2f*)(uP + 4 * k);
      b[k] = *(const v2f*)(vP + 4 * k);
    }

    v8f acc = {};
#pragma unroll
    for (int k = 0; k < 16; ++k) {
      // (neg_a, A, neg_b, B, c_mod, C, reuse_a, reuse_b)
      acc = __builtin_amdgcn_wmma_f32_16x16x4_f32(
          false, a[k], false, b[k], (short)0, acc, false, false);
    }

    // Diagonal extraction from the 16x16 f32 C/D layout:
    //   i in 0..7  -> lane i,     component i
    //   i in 8..15 -> lane i+16,  component i-8
    const int sel = lane & 7;
    float val = acc[0];
    val = (sel == 1) ? acc[1] : val;
    val = (sel == 2) ? acc[2] : val;
    val = (sel == 3) ? acc[3] : val;
    val = (sel == 4) ? acc[4] : val;
    val = (sel == 5) ? acc[5] : val;
    val = (sel == 6) ? acc[6] : val;
    val = (sel == 7) ? acc[7] : val;

    const bool lo = (lane < 8);
    const bool hi = (lane >= 24);
    const int i = lo ? lane : (lane - 16);
    if (lo || hi) out[edge_base + i] = val;
  } else {
    // ---- tail path (never taken for E=1,280,000; wave-uniform branch) ----
    if (lane < 16) {
      const int e = edge_base + lane;
      if (e < n_edges) {
        const float* up = x + (size_t)src[e] * 64;
        const float* vp = x + (size_t)dst[e] * 64;
        float s = 0.f;
#pragma unroll 8
        for (int d = 0; d < 64; ++d) s += up[d] * vp[d];
        out[e] = s;
      }
    }
  }
}

extern "C" void kernel_launch(void* const* d_in, const int* in_sizes, int n_in,
                              void* d_out, int out_size, void* d_ws, size_t ws_size,
                              hipStream_t stream) {
  (void)n_in; (void)out_size; (void)d_ws; (void)ws_size;
  const float* x = (const float*)d_in[0];
  const long long* src = (const long long*)d_in[1];
  const long long* dst = (const long long*)d_in[2];
  float* out = (float*)d_out;
  const int n_edges = in_sizes[1];

  const int block = 256;                       // 8 waves -> 128 edges/block
  const int grid = (n_edges + 127) / 128;
  edge_dot_wmma_kernel<<<grid, block, 0, stream>>>(x, src, dst, out, n_edges);
}